// MDLLoss_5909874999671
// MI455X (gfx1250) — compile-verified
//
#include <hip/hip_runtime.h>
#include <math.h>

// ---------------- problem geometry (reference: B=65536, F=256) --------------
#define F           256            // features == blockDim.x (one feature/thread)
#define RTILE       32             // rows per LDS tile  -> 32*256*4 = 32 KB
#define TILE_ELEMS  (RTILE * F)    // floats per tile
#define CHUNKS      (RTILE / 4)    // 16B async-DMA chunks per thread per tile (8)

// ---------------- CDNA5 async global->LDS path (guarded) --------------------
#if defined(__AMDGCN__) && __has_builtin(__builtin_amdgcn_global_load_async_to_lds_b128)
  #define HAS_ASYNC 1
#else
  #define HAS_ASYNC 0
#endif

#if defined(__AMDGCN__)
typedef int v4i __attribute__((ext_vector_type(4)));
#define AS1 __attribute__((address_space(1)))
#define AS3 __attribute__((address_space(3)))
#endif

#if HAS_ASYNC
  #if __has_builtin(__builtin_amdgcn_s_wait_asynccnt)
    #define WAIT_ASYNC(n) __builtin_amdgcn_s_wait_asynccnt(n)
  #else
    #define WAIT_ASYNC(n) asm volatile("s_wait_asynccnt %0" :: "i"(n) : "memory")
  #endif
#else
  #define WAIT_ASYNC(n) do {} while (0)
#endif

// Issue one 32-row tile: 256 threads x 8 x b128 = 32 KB, coalesced memcpy.
__device__ __forceinline__ void issue_tile(const float* __restrict__ gbase,
                                           float* lbase, int t) {
#pragma unroll
  for (int j = 0; j < CHUNKS; ++j) {
    const int c = (j * F + t) * 4;            // float index, 16B aligned
#if HAS_ASYNC
    __builtin_amdgcn_global_load_async_to_lds_b128(
        (AS1 v4i*)(gbase + c),
        (AS3 v4i*)(lbase + c),
        /*offset=*/0, /*cpol=*/0);
#else
    *(float4*)(lbase + c) = *(const float4*)(gbase + c);
#endif
  }
}

// ---------------- per-element Yeo-Johnson (mirrors reference masks) ---------
__device__ __forceinline__ void accum_elem(float x, float lam1, float lam2,
                                           float& cnt, float& s1, float& s2,
                                           float& sj) {
  const float EPS = 1e-8f;
  float y = 0.f, lj = 0.f, fin = 0.f;
  if (x >= 0.f) {                                   // safe_pos branch
    if (x <= 1000.f && fabsf(lam1) <= 1.9f) {
      float t = __logf(1.0f + x);                   // ln(1+x)
      float yp = (fabsf(lam1) < EPS) ? t
               : (__expf(lam1 * t) - 1.0f) / lam1;  // ((1+x)^l1 - 1)/l1
      y = yp; lj = (lam1 - 1.0f) * t; fin = 1.0f;
    }
  } else if (x > -0.99f && fabsf(lam2) <= 1.9f) {   // safe_neg branch
    float xn = fmaxf(x, -0.989f);                   // clip(x, -0.989, 0)
    float t  = __logf(1.0f - xn);                   // ln(1-xn)
    float l2 = 2.0f - lam2;
    float yn = (fabsf(l2) < EPS) ? -t
             : -(__expf(l2 * t) - 1.0f) / l2;
    y = yn; lj = (1.0f - lam2) * t; fin = 1.0f;
  }
  cnt += fin;
  s1  += y;
  s2   = fmaf(y, y, s2);
  sj  += lj;
}

// ---------------- pass 1: streaming column reduction ------------------------
// ws layout: float[4][nBlocks][F]  (q = 0:cnt 1:sum 2:sumsq 3:sum_log_jac)
__global__ __launch_bounds__(F) void mdl_pass1(
    const float* __restrict__ res, const float* __restrict__ l1v,
    const float* __restrict__ l2v, float* __restrict__ ws,
    int rows, int nTiles, int nBlocks) {
  __shared__ float smem[2 * TILE_ELEMS];            // 64 KB double buffer
  const int t = threadIdx.x;
  const float lam1 = l1v[t];
  const float lam2 = l2v[t];
  float cnt = 0.f, s1 = 0.f, s2 = 0.f, sj = 0.f;

  int tile = blockIdx.x;
  if (tile < nTiles)
    issue_tile(res + (size_t)tile * TILE_ELEMS, smem, t);   // prologue -> buf0

  int buf = 0;
  for (; tile < nTiles; tile += nBlocks) {
    const int nxt = tile + nBlocks;
    if (nxt < nTiles) {
      issue_tile(res + (size_t)nxt * TILE_ELEMS, smem + (buf ^ 1) * TILE_ELEMS, t);
      WAIT_ASYNC(CHUNKS);   // in-order: current tile's 8 DMAs are done
    } else {
      WAIT_ASYNC(0);
    }
    __syncthreads();        // all waves' DMAs for current buffer landed

    const float* p = smem + buf * TILE_ELEMS;
#pragma unroll 8
    for (int r = 0; r < RTILE; ++r)
      accum_elem(p[r * F + t], lam1, lam2, cnt, s1, s2, sj); // conflict-free DS reads

    __syncthreads();        // done reading before this buffer is re-issued
    buf ^= 1;
  }

  // ragged tail (rows not multiple of RTILE) — straight from global, block 0
  if (blockIdx.x == 0)
    for (int r = nTiles * RTILE; r < rows; ++r)
      accum_elem(res[(size_t)r * F + t], lam1, lam2, cnt, s1, s2, sj);

  const size_t bi = blockIdx.x;
  ws[(0 * (size_t)nBlocks + bi) * F + t] = cnt;
  ws[(1 * (size_t)nBlocks + bi) * F + t] = s1;
  ws[(2 * (size_t)nBlocks + bi) * F + t] = s2;
  ws[(3 * (size_t)nBlocks + bi) * F + t] = sj;
}

// ---------------- pass 2: per-feature bits + final scalar -------------------
__global__ __launch_bounds__(F) void mdl_pass2(
    const float* __restrict__ ws, const float* __restrict__ resolutions,
    float* __restrict__ out, int nBlocks, int rows) {
  const int f = threadIdx.x;
  double cnt = 0.0, s1 = 0.0, s2 = 0.0, sj = 0.0;
  for (int b = 0; b < nBlocks; ++b) {
    cnt += ws[(0 * (size_t)nBlocks + b) * F + f];
    s1  += ws[(1 * (size_t)nBlocks + b) * F + f];
    s2  += ws[(2 * (size_t)nBlocks + b) * F + f];
    sj  += ws[(3 * (size_t)nBlocks + b) * F + f];
  }
  const double Bd     = (double)rows;
  const double n_exc  = Bd - cnt;
  const double denom  = fmax(cnt, 1.0);
  const double mean   = s1 / denom;
  double var = (s2 - 2.0 * mean * s1 + mean * mean * cnt) / denom;
  var = fmax(var, 1e-12);

  const double LOG2_2PIE = 2.0470955851806411;   // 0.5*log2(2*pi*e)
  const double LN2       = 0.6931471805599453;
  const double LAMBITS   = 13.287712379549449;   // 2*log2(100)

  double diff_bits = (cnt > 1.0) ? cnt * (LOG2_2PIE + 0.5 * log2(var)) : 0.0;
  double jac_bits  = sj / LN2;
  double exc_bits  = (n_exc > 0.0) ? n_exc * (-log2((double)resolutions[f])) : 0.0;
  double lb        = lgamma(Bd + 1.0) - lgamma(n_exc + 1.0) - lgamma(Bd - n_exc + 1.0);
  double part_bits = (n_exc > 0.0 && n_exc < Bd) ? lb / LN2 : 0.0;

  double bits = diff_bits + jac_bits + exc_bits + part_bits + LAMBITS;

  __shared__ double red[F];
  red[f] = bits;
  __syncthreads();
  for (int s = F / 2; s > 0; s >>= 1) {
    if (f < s) red[f] += red[f + s];
    __syncthreads();
  }
  if (f == 0) out[0] = (float)red[0];
}

// ---------------- launch ----------------------------------------------------
extern "C" void kernel_launch(void* const* d_in, const int* in_sizes, int n_in,
                              void* d_out, int out_size, void* d_ws, size_t ws_size,
                              hipStream_t stream) {
  (void)n_in; (void)out_size;
  const float* residuals   = (const float*)d_in[0];
  const float* lambda1     = (const float*)d_in[1];
  const float* lambda2     = (const float*)d_in[2];
  const float* resolutions = (const float*)d_in[3];

  const int rows   = in_sizes[0] / F;       // 65536
  const int nTiles = rows / RTILE;          // 2048

  int nBlocks = 512;                        // 4096 waves; ~16 MB DMA in flight
  const size_t perBlock = 4 * (size_t)F * sizeof(float);
  if (ws_size < (size_t)nBlocks * perBlock) {
    nBlocks = (int)(ws_size / perBlock);
    if (nBlocks < 1) nBlocks = 1;
  }

  mdl_pass1<<<nBlocks, F, 0, stream>>>(residuals, lambda1, lambda2,
                                       (float*)d_ws, rows, nTiles, nBlocks);
  mdl_pass2<<<1, F, 0, stream>>>((const float*)d_ws, resolutions,
                                 (float*)d_out, nBlocks, rows);
}